// DescriptionEncoder_77300821393909
// MI455X (gfx1250) — compile-verified
//
#include <hip/hip_runtime.h>
#include <hip/hip_bf16.h>

#define SEQ 8192
#define EMB 50
#define HID 100
#define G4  400   // 4*HID

typedef float v2f __attribute__((ext_vector_type(2)));
typedef float v8f __attribute__((ext_vector_type(8)));

// ---------------------------------------------------------------------------
// Kernel A: xg[s][j*4+g] = sum_k emb[x[s]][k] * w_ih[g*100+j][k] + bias
// One wave per 16x16 output tile, fp32 WMMA 16x16x4, gather fused into A frag.
// Output stored PERMUTED (gate-interleaved) so the scan reads coalesced.
// ---------------------------------------------------------------------------
__global__ __launch_bounds__(128)
void xg_gemm_kernel(const int* __restrict__ x, const float* __restrict__ emb,
                    const float* __restrict__ w_ih, const float* __restrict__ b_ih,
                    const float* __restrict__ b_hh, float* __restrict__ xg) {
    const int lane  = threadIdx.x & 31;
    const int wave  = threadIdx.x >> 5;
    const int tile  = blockIdx.x * 4 + wave;
    const int NT_N  = G4 / 16;                 // 25 tiles along gate dim
    const int tm    = (tile / NT_N) * 16;      // seq-tile base (0..8176)
    const int tn    = (tile % NT_N) * 16;      // gate-tile base
    const int l15   = lane & 15;
    const int khalf = lane >> 4;               // K split across half-waves

    // gather index for this lane's A row
    const int       s    = tm + l15;
    const long long arow = (long long)x[s] * EMB;

    // B column / bias column for this lane
    const int       ncol = tn + l15;
    const long long brow = (long long)ncol * EMB;
    const float     bias = b_ih[ncol] + b_hh[ncol];

    v8f c;
#pragma unroll
    for (int r = 0; r < 8; ++r) c[r] = bias;   // accumulate on top of bias

    for (int k0 = 0; k0 < EMB; k0 += 4) {      // 13 iters, last padded
        const int ka  = k0 + 2 * khalf;
        // branchless tail: clamp address, cndmask the value
        const int ka0 = (ka     < EMB) ? ka     : 0;
        const int ka1 = (ka + 1 < EMB) ? ka + 1 : 0;
        v2f a, b;
        float va0 = emb[arow + ka0],  va1 = emb[arow + ka1];
        float vb0 = w_ih[brow + ka0], vb1 = w_ih[brow + ka1];
        a[0] = (ka     < EMB) ? va0 : 0.0f;
        a[1] = (ka + 1 < EMB) ? va1 : 0.0f;
        b[0] = (ka     < EMB) ? vb0 : 0.0f;    // B[k][n] = w_ih[n][k]
        b[1] = (ka + 1 < EMB) ? vb1 : 0.0f;
        c = __builtin_amdgcn_wmma_f32_16x16x4_f32(false, a, false, b,
                                                  (short)0, c, false, false);
    }

    // permuted store: col c in [0,400) -> (c%100)*4 + c/100  (j*4 + gate)
    const int  cc = tn + l15;
    const int  cp = (cc % HID) * 4 + (cc / HID);
#pragma unroll
    for (int r = 0; r < 8; ++r) {
        const int row = tm + r + 8 * khalf;    // C layout: VGPR r -> M=r / M=r+8
        xg[(long long)row * G4 + cp] = c[r];
    }
}

// ---------------------------------------------------------------------------
// Kernel B: serial LSTM scan. Single workgroup on one WGP.
// tid = j*4 + gate: the 4 gates of output j sit in 4 adjacent lanes of ONE
// wave -> gather with __shfl (ds_bpermute), no barrier. h double-buffered in
// LDS -> exactly one s_barrier per step. xg load software-pipelined.
// ---------------------------------------------------------------------------
__device__ __forceinline__ float fast_sigmoid(float v) {
    return 1.0f / (1.0f + __expf(-v));
}
__device__ __forceinline__ float fast_tanh(float v) {
    return 1.0f - 2.0f / (__expf(2.0f * v) + 1.0f);
}

__global__ __launch_bounds__(416)
void lstm_scan_kernel(const float* __restrict__ xg, const float* __restrict__ w_hh,
                      float* __restrict__ hs) {
    __shared__ __align__(16) float sh[2][112];   // double-buffered h
    const int tid  = threadIdx.x;
    const int lane = tid & 31;
    const int gt   = tid & 3;        // gate: 0=i 1=f 2=g 3=o
    const int j    = tid >> 2;       // output index 0..99

    float w[HID];                    // w_hh row (gt*100 + j), kept in VGPRs
    if (tid < G4) {
        const long long r = (long long)(gt * HID + j) * HID;
#pragma unroll
        for (int k = 0; k < HID; ++k) w[k] = w_hh[r + k];
    }
    float cst = 0.0f;
    if (tid < HID) { sh[0][tid] = 0.0f; sh[1][tid] = 0.0f; }
    __syncthreads();

    float xv = (tid < G4) ? xg[tid] : 0.0f;      // pipelined xg value for t=0

    for (int t = 0; t < SEQ; ++t) {
        const int p = t & 1;
        if (tid < G4) {
            // issue next step's xg load early (independent of LDS)
            const float xn = (t + 1 < SEQ) ? xg[(long long)(t + 1) * G4 + tid]
                                           : 0.0f;
            const float4* h4p = (const float4*)sh[p];
            float a0 = 0.f, a1 = 0.f, a2 = 0.f, a3 = 0.f;
#pragma unroll
            for (int q = 0; q < HID / 4; ++q) {  // 25 x ds_load_b128 broadcast
                float4 h4 = h4p[q];
                a0 += w[4 * q + 0] * h4.x;
                a1 += w[4 * q + 1] * h4.y;
                a2 += w[4 * q + 2] * h4.z;
                a3 += w[4 * q + 3] * h4.w;
            }
            const float acc = xv + ((a0 + a1) + (a2 + a3));
            // gather the 4 gate pre-activations within the wave
            const int base = lane & ~3;
            const float gi = __shfl(acc, base + 0);
            const float gf = __shfl(acc, base + 1);
            const float gg = __shfl(acc, base + 2);
            const float go = __shfl(acc, base + 3);
            if (gt == 0) {
                const float i = fast_sigmoid(gi);
                const float f = fast_sigmoid(gf);
                const float g = fast_tanh(gg);
                const float o = fast_sigmoid(go);
                cst = f * cst + i * g;
                const float h = o * fast_tanh(cst);
                sh[1 - p][j] = h;
                hs[(long long)t * HID + j] = h;
            }
            xv = xn;
        }
        __syncthreads();             // h_t ready; also closes WAR on buffers
    }
}

// ---------------------------------------------------------------------------
// Kernel C: out = hs @ fc_w.T + fc_b   (M=8192, N=100 -> 7 tiles, K=100)
// ---------------------------------------------------------------------------
__global__ __launch_bounds__(128)
void fc_gemm_kernel(const float* __restrict__ hs, const float* __restrict__ fc_w,
                    const float* __restrict__ fc_b, float* __restrict__ out) {
    const int lane  = threadIdx.x & 31;
    const int wave  = threadIdx.x >> 5;
    const int NT_N  = 7;                       // ceil(100/16)
    const int tile  = blockIdx.x * 4 + wave;
    const int tm    = (tile / NT_N) * 16;
    const int tn    = (tile % NT_N) * 16;
    const int l15   = lane & 15;
    const int khalf = lane >> 4;

    const int  ncol   = tn + l15;
    const bool nvalid = (ncol < HID);
    const long long arow = (long long)(tm + l15) * HID;
    const long long brow = (long long)(nvalid ? ncol : 0) * HID;
    const float bias  = nvalid ? fc_b[nvalid ? ncol : 0] : 0.0f;

    v8f c;
#pragma unroll
    for (int r = 0; r < 8; ++r) c[r] = bias;

    for (int k0 = 0; k0 < HID; k0 += 4) {      // 25 exact iterations
        const int ka = k0 + 2 * khalf;
        v2f a, b;
        a[0] = hs[arow + ka];
        a[1] = hs[arow + ka + 1];
        float vb0 = fc_w[brow + ka], vb1 = fc_w[brow + ka + 1];
        b[0] = nvalid ? vb0 : 0.0f;            // branchless: cndmask, not exec
        b[1] = nvalid ? vb1 : 0.0f;
        c = __builtin_amdgcn_wmma_f32_16x16x4_f32(false, a, false, b,
                                                  (short)0, c, false, false);
    }

#pragma unroll
    for (int r = 0; r < 8; ++r) {
        const int row = tm + r + 8 * khalf;
        const int col = tn + l15;
        if (col < HID) out[(long long)row * HID + col] = c[r];
    }
}

// ---------------------------------------------------------------------------
extern "C" void kernel_launch(void* const* d_in, const int* in_sizes, int n_in,
                              void* d_out, int out_size, void* d_ws, size_t ws_size,
                              hipStream_t stream) {
    const int*   x    = (const int*)  d_in[0];
    const float* emb  = (const float*)d_in[1];
    const float* w_ih = (const float*)d_in[2];
    const float* w_hh = (const float*)d_in[3];
    const float* b_ih = (const float*)d_in[4];
    const float* b_hh = (const float*)d_in[5];
    const float* fc_w = (const float*)d_in[6];
    const float* fc_b = (const float*)d_in[7];
    float* out = (float*)d_out;

    float* xg = (float*)d_ws;                         // SEQ*G4  = 13.1 MB (permuted)
    float* hs = xg + (size_t)SEQ * G4;                // SEQ*HID =  3.3 MB

    // Phase 1: gather + input GEMM (512 m-tiles x 25 n-tiles, 4 waves/block)
    xg_gemm_kernel<<<(512 * 25) / 4, 128, 0, stream>>>(x, emb, w_ih, b_ih, b_hh, xg);
    // Phase 2: serial scan, one WGP, one barrier per step
    lstm_scan_kernel<<<1, 416, 0, stream>>>(xg, w_hh, hs);
    // Phase 3: output projection (512 x 7 tiles)
    fc_gemm_kernel<<<(512 * 7) / 4, 128, 0, stream>>>(hs, fc_w, fc_b, out);
}